// ContrastiveM3LDecoderNetwork_83004537963096
// MI455X (gfx1250) — compile-verified
//
#include <hip/hip_runtime.h>
#include <hip/hip_bf16.h>
#include <stdint.h>

// ---------------------------------------------------------------------------
// Problem dims
// ---------------------------------------------------------------------------
#define NB   256      // batch
#define NL   2        // languages
#define NK   100      // topics
#define NV   50000    // vocab
#define NH   100      // hidden
#define NBERT 768
#define NIMG  2048
#define EPS  1e-5f
#define HPAD 112      // NH padded to 7 WMMA tiles

typedef __attribute__((ext_vector_type(2))) float v2f;
typedef __attribute__((ext_vector_type(8))) float v8f;

// ---------------------------------------------------------------------------
// Workspace layout (float element offsets)
// ---------------------------------------------------------------------------
#define WS_H1     0              // 3 * 256 * 112
#define WS_H2     86016
#define WS_MU     172032
#define WS_LS     258048
#define WS_Z      344064         // 3 * 256 * 100
#define WS_THETA  420864         // 3 * 256 * 100
#define WS_MEAN   497664         // 2 * 50000
#define WS_RSTD   597664         // 2 * 50000
#define WS_LOGITS 697664         // 2 * 256 * 50000

// Output layout (float element offsets, reference return order)
#define OUT_PM    0
#define OUT_PV    100
#define OUT_NETB  200            // per-net block: mu | exp(ls) | ls, each 25600
#define OUT_NETSTRIDE 76800
#define OUT_WORD  230600         // 2*256*50000
#define OUT_THETA 25830600       // 3*256*100

// ---------------------------------------------------------------------------
// Deterministic hash-based standard normal (Box-Muller)
// ---------------------------------------------------------------------------
__device__ __forceinline__ float hash_u01(uint32_t x) {
    x ^= x >> 17; x *= 0xed5ad4bbu;
    x ^= x >> 11; x *= 0xac4c1b51u;
    x ^= x >> 15; x *= 0x31848babu;
    x ^= x >> 14;
    return (float)(x >> 8) * (1.0f / 16777216.0f) + (0.5f / 16777216.0f);
}
__device__ __forceinline__ float gauss_hash(uint32_t idx) {
    float u1 = hash_u01(idx * 2654435761u + 0x9E3779B9u);
    float u2 = hash_u01(idx * 2246822519u + 0x85EBCA6Bu);
    float r  = sqrtf(-2.0f * __logf(fmaxf(u1, 1e-12f)));
    return r * __cosf(6.28318530718f * u2);
}

// ---------------------------------------------------------------------------
// Generic fp32 WMMA GEMM: C[16m x 16n tile] = act(A @ W + bias)
// A: M x K, row stride lda.  W: K x N row-major (ld = N).  C: row stride ldc
// grid = (ceil(N/16), M/16), block = 32 (one wave)
// act: 0 = identity, 1 = softplus
// Column guard is clamp+mask (no EXEC divergence around the WMMAs).
// ---------------------------------------------------------------------------
__global__ void gemm_bias_act_wmma(const float* __restrict__ A, int lda,
                                   const float* __restrict__ W,
                                   const float* __restrict__ bias,
                                   float* __restrict__ C, int ldc,
                                   int N, int K, int act)
{
    const int lane = threadIdx.x & 31;
    const int half = lane >> 4;        // K-pair selector (0: k0/k1, 1: k2/k3)
    const int lid  = lane & 15;
    const int n0   = blockIdx.x * 16;
    const int m0   = blockIdx.y * 16;

    const float* arow = A + (size_t)(m0 + lid) * lda;
    const int   ncol  = n0 + lid;
    const int   nc    = (ncol < N) ? ncol : (N - 1);   // clamped, always in-bounds
    const float nokf  = (ncol < N) ? 1.0f : 0.0f;      // mask

    const float* wp = W + nc;

    v8f acc = {};
    for (int kb = 0; kb < K; kb += 4) {
        const int ka = kb + 2 * half;
        v2f a, b;
        a.x = arow[ka];
        a.y = arow[ka + 1];
        b.x = wp[(size_t)ka * N] * nokf;
        b.y = wp[(size_t)(ka + 1) * N] * nokf;
        acc = __builtin_amdgcn_wmma_f32_16x16x4_f32(false, a, false, b,
                                                    (short)0, acc, false, false);
    }
    const float bv = bias[nc] * nokf;
    float* cp = C + (size_t)(m0 + half * 8) * ldc + ncol;   // ncol < HPAD always
#pragma unroll
    for (int r = 0; r < 8; ++r) {
        float v = acc[r] + bv;
        if (act) v = (v > 20.0f) ? v : log1pf(__expf(v));
        cp[(size_t)r * ldc] = v;
    }
}

// ---------------------------------------------------------------------------
// Decode GEMM: logits[l][b][v] = theta[l][b][:] @ beta[l][:][v]
// One wave: 16 rows x 80 cols (5 WMMA N-tiles share one A fragment).
// A fragments (25 x v2f = 50 VGPRs) preloaded; K loop fully unrolled.
// block = 128 (4 waves, different M-tiles, same columns -> B reuse in L0)
// grid = (50000/80 = 625, 16/4 = 4, NL = 2)
// ---------------------------------------------------------------------------
#define DNT 5
#define KSTEPS 25     // NK / 4
__global__ void decode_gemm_wmma(const float* __restrict__ theta,  // [2][256][100]
                                 const float* __restrict__ beta,   // [2][100][50000]
                                 float* __restrict__ logits)       // [2][256][50000]
{
    const int l    = blockIdx.z;
    const int lane = threadIdx.x & 31;
    const int wave = threadIdx.x >> 5;
    const int half = lane >> 4;
    const int lid  = lane & 15;
    const int m0   = (blockIdx.y * 4 + wave) * 16;
    const int n0   = blockIdx.x * (16 * DNT);

    const float* arow = theta + ((size_t)l * NB + m0 + lid) * NK + 2 * half;
    const float* Bp   = beta + (size_t)l * NK * NV + 2 * half * NV + n0 + lid;

    // Preload the entire A tile for this wave (loop-invariant across N-tiles)
    v2f av[KSTEPS];
#pragma unroll
    for (int i = 0; i < KSTEPS; ++i) {
        av[i].x = arow[4 * i];
        av[i].y = arow[4 * i + 1];
    }

    v8f acc[DNT] = {};
#pragma unroll
    for (int i = 0; i < KSTEPS; ++i) {
        const float* b0 = Bp + (size_t)(4 * i) * NV;
#pragma unroll
        for (int t = 0; t < DNT; ++t) {
            v2f b;
            b.x = b0[t * 16];
            b.y = b0[t * 16 + NV];
            acc[t] = __builtin_amdgcn_wmma_f32_16x16x4_f32(false, av[i], false, b,
                                                           (short)0, acc[t], false, false);
        }
    }

    float* cp = logits + ((size_t)l * NB + m0 + half * 8) * NV + n0 + lid;
#pragma unroll
    for (int t = 0; t < DNT; ++t)
#pragma unroll
        for (int r = 0; r < 8; ++r)
            cp[(size_t)r * NV + t * 16] = acc[t][r];
}

// ---------------------------------------------------------------------------
// BatchNorm over batch + write mu/exp(ls)/ls + reparameterized sample z
// grid = (100 cols, 3 nets), block = 256 (one thread per batch row)
// ---------------------------------------------------------------------------
__device__ __forceinline__ float block_sum_256(float v, float* sh) {
    const int t = threadIdx.x;
    sh[t] = v; __syncthreads();
    for (int s = 128; s > 0; s >>= 1) {
        if (t < s) sh[t] += sh[t + s];
        __syncthreads();
    }
    float r = sh[0]; __syncthreads();
    return r;
}

__global__ void bn_sample_kernel(const float* __restrict__ mu_raw,  // [3][256][112]
                                 const float* __restrict__ ls_raw,  // [3][256][112]
                                 float* __restrict__ out,
                                 float* __restrict__ zbuf)          // [3][256][100]
{
    __shared__ float sh[256];
    const int net = blockIdx.y;
    const int col = blockIdx.x;
    const int b   = threadIdx.x;

    const size_t base = ((size_t)net * NB + b) * HPAD + col;
    const float mv = mu_raw[base];
    const float lv = ls_raw[base];

    const float s_mu  = block_sum_256(mv, sh);
    const float ss_mu = block_sum_256(mv * mv, sh);
    const float s_ls  = block_sum_256(lv, sh);
    const float ss_ls = block_sum_256(lv * lv, sh);

    const float m_mu = s_mu * (1.0f / NB);
    const float m_ls = s_ls * (1.0f / NB);
    const float v_mu = fmaxf(ss_mu * (1.0f / NB) - m_mu * m_mu, 0.0f);
    const float v_ls = fmaxf(ss_ls * (1.0f / NB) - m_ls * m_ls, 0.0f);

    const float mu_n = (mv - m_mu) * rsqrtf(v_mu + EPS);
    const float ls_n = (lv - m_ls) * rsqrtf(v_ls + EPS);

    float* ob = out + OUT_NETB + (size_t)net * OUT_NETSTRIDE + (size_t)b * NK + col;
    ob[0]     = mu_n;
    ob[25600] = __expf(ls_n);
    ob[51200] = ls_n;

    const uint32_t idx = (uint32_t)(net * NB * NK + b * NK + col);
    zbuf[idx] = mu_n + gauss_hash(idx) * __expf(0.5f * ls_n);
}

// ---------------------------------------------------------------------------
// Softmax over K=100 per (net, b) row -> thetas (to d_out and to workspace)
// grid = 768, block = 128
// ---------------------------------------------------------------------------
__global__ void theta_softmax_kernel(const float* __restrict__ z,
                                     float* __restrict__ theta_ws,
                                     float* __restrict__ theta_out)
{
    __shared__ float sh[128];
    const int row = blockIdx.x;
    const int t   = threadIdx.x;
    const float* zr = z + (size_t)row * NK;

    float mx = -1e30f;
    for (int k = t; k < NK; k += 128) mx = fmaxf(mx, zr[k]);
    sh[t] = mx; __syncthreads();
    for (int s = 64; s > 0; s >>= 1) {
        if (t < s) sh[t] = fmaxf(sh[t], sh[t + s]);
        __syncthreads();
    }
    mx = sh[0]; __syncthreads();

    float sm = 0.0f;
    for (int k = t; k < NK; k += 128) sm += __expf(zr[k] - mx);
    sh[t] = sm; __syncthreads();
    for (int s = 64; s > 0; s >>= 1) {
        if (t < s) sh[t] += sh[t + s];
        __syncthreads();
    }
    const float inv = 1.0f / sh[0];

    for (int k = t; k < NK; k += 128) {
        const float v = __expf(zr[k] - mx) * inv;
        theta_ws[(size_t)row * NK + k]  = v;
        theta_out[(size_t)row * NK + k] = v;
    }
}

// ---------------------------------------------------------------------------
// Per-(l, v) batch statistics over b: mean & rstd
// grid = (196, 2), block = 256, one thread per column
// ---------------------------------------------------------------------------
__global__ void col_stats_kernel(const float* __restrict__ logits,
                                 float* __restrict__ mean,
                                 float* __restrict__ rstd)
{
    const int l = blockIdx.y;
    const int v = blockIdx.x * 256 + threadIdx.x;
    if (v >= NV) return;
    const float* p = logits + (size_t)l * NB * NV + v;
    float s = 0.0f, ss = 0.0f;
    for (int b = 0; b < NB; ++b) {
        const float x = p[(size_t)b * NV];
        s += x; ss += x * x;
    }
    const float m   = s * (1.0f / NB);
    const float var = fmaxf(ss * (1.0f / NB) - m * m, 0.0f);
    mean[(size_t)l * NV + v] = m;
    rstd[(size_t)l * NV + v] = rsqrtf(var + EPS);
}

// ---------------------------------------------------------------------------
// Row softmax over V=50000 with fused batchnorm normalize (online max/sum)
// grid = 512 (l*256+b), block = 256
// ---------------------------------------------------------------------------
__global__ void row_softmax_kernel(const float* __restrict__ logits,
                                   const float* __restrict__ mean,
                                   const float* __restrict__ rstd,
                                   float* __restrict__ word_out)
{
    __shared__ float smx[256], ssm[256];
    const int row = blockIdx.x;
    const int l   = row >> 8;
    const int t   = threadIdx.x;
    const float* p  = logits + (size_t)row * NV;
    const float* mm = mean + (size_t)l * NV;
    const float* rr = rstd + (size_t)l * NV;

    float mx = -1e30f, sm = 0.0f;
    for (int v = t; v < NV; v += 256) {
        const float x = (p[v] - mm[v]) * rr[v];
        if (x > mx) { sm = sm * __expf(mx - x) + 1.0f; mx = x; }
        else        { sm += __expf(x - mx); }
    }
    smx[t] = mx; ssm[t] = sm; __syncthreads();
    for (int s = 128; s > 0; s >>= 1) {
        if (t < s) {
            const float m2 = smx[t + s], s2 = ssm[t + s];
            const float M  = fmaxf(smx[t], m2);
            ssm[t] = ssm[t] * __expf(smx[t] - M) + s2 * __expf(m2 - M);
            smx[t] = M;
        }
        __syncthreads();
    }
    const float M = smx[0];
    const float invS = 1.0f / ssm[0];

    float* o = word_out + (size_t)row * NV;
    for (int v = t; v < NV; v += 256) {
        const float x = (p[v] - mm[v]) * rr[v];
        o[v] = __expf(x - M) * invS;
    }
}

// ---------------------------------------------------------------------------
// Passthrough priors
// ---------------------------------------------------------------------------
__global__ void copy_prior_kernel(const float* __restrict__ pm,
                                  const float* __restrict__ pv,
                                  float* __restrict__ out)
{
    const int t = threadIdx.x;
    if (t < 100)       out[OUT_PM + t]       = pm[t];
    else if (t < 200)  out[OUT_PV + t - 100] = pv[t - 100];
}

// ---------------------------------------------------------------------------
// Launch
// ---------------------------------------------------------------------------
extern "C" void kernel_launch(void* const* d_in, const int* in_sizes, int n_in,
                              void* d_out, int out_size, void* d_ws, size_t ws_size,
                              hipStream_t stream)
{
    (void)in_sizes; (void)n_in; (void)out_size; (void)ws_size;

    const float* x_bert  = (const float*)d_in[1];   // [256][2][768]
    const float* x_image = (const float*)d_in[2];   // [256][2048]
    const float* pm      = (const float*)d_in[3];
    const float* pv      = (const float*)d_in[4];
    const float* beta    = (const float*)d_in[5];   // [2][100][50000]

    // per-net weights: Wa, ba, Wh, bh, Wmu, bmu, Wsig, bsig starting at 6
    const float* Wa[3]; const float* ba[3]; const float* Wh[3]; const float* bh[3];
    const float* Wmu[3]; const float* bmu[3]; const float* Wsig[3]; const float* bsig[3];
    for (int t = 0; t < 3; ++t) {
        Wa[t]   = (const float*)d_in[6 + 8 * t + 0];
        ba[t]   = (const float*)d_in[6 + 8 * t + 1];
        Wh[t]   = (const float*)d_in[6 + 8 * t + 2];
        bh[t]   = (const float*)d_in[6 + 8 * t + 3];
        Wmu[t]  = (const float*)d_in[6 + 8 * t + 4];
        bmu[t]  = (const float*)d_in[6 + 8 * t + 5];
        Wsig[t] = (const float*)d_in[6 + 8 * t + 6];
        bsig[t] = (const float*)d_in[6 + 8 * t + 7];
    }

    float* out = (float*)d_out;
    float* ws  = (float*)d_ws;

    float* h1     = ws + WS_H1;
    float* h2     = ws + WS_H2;
    float* mu_raw = ws + WS_MU;
    float* ls_raw = ws + WS_LS;
    float* zbuf   = ws + WS_Z;
    float* th_ws  = ws + WS_THETA;
    float* meanb  = ws + WS_MEAN;
    float* rstdb  = ws + WS_RSTD;
    float* logits = ws + WS_LOGITS;

    hipLaunchKernelGGL(copy_prior_kernel, dim3(1), dim3(256), 0, stream, pm, pv, out);

    const dim3 gG(7, 16), bG(32);
    const float* Ain[3]  = { x_bert, x_bert + NBERT, x_image };
    const int    ldA[3]  = { 2 * NBERT, 2 * NBERT, NIMG };
    const int    Kin[3]  = { NBERT, NBERT, NIMG };

    for (int t = 0; t < 3; ++t) {
        float* h1t = h1 + (size_t)t * NB * HPAD;
        float* h2t = h2 + (size_t)t * NB * HPAD;
        float* mut = mu_raw + (size_t)t * NB * HPAD;
        float* lst = ls_raw + (size_t)t * NB * HPAD;

        hipLaunchKernelGGL(gemm_bias_act_wmma, gG, bG, 0, stream,
                           Ain[t], ldA[t], Wa[t], ba[t], h1t, HPAD, NH, Kin[t], 1);
        hipLaunchKernelGGL(gemm_bias_act_wmma, gG, bG, 0, stream,
                           h1t, HPAD, Wh[t], bh[t], h2t, HPAD, NH, NH, 1);
        hipLaunchKernelGGL(gemm_bias_act_wmma, gG, bG, 0, stream,
                           h2t, HPAD, Wmu[t], bmu[t], mut, HPAD, NK, NH, 0);
        hipLaunchKernelGGL(gemm_bias_act_wmma, gG, bG, 0, stream,
                           h2t, HPAD, Wsig[t], bsig[t], lst, HPAD, NK, NH, 0);
    }

    hipLaunchKernelGGL(bn_sample_kernel, dim3(NK, 3), dim3(256), 0, stream,
                       mu_raw, ls_raw, out, zbuf);

    hipLaunchKernelGGL(theta_softmax_kernel, dim3(3 * NB), dim3(128), 0, stream,
                       zbuf, th_ws, out + OUT_THETA);

    hipLaunchKernelGGL(decode_gemm_wmma, dim3(NV / (16 * DNT), 4, NL), dim3(128), 0, stream,
                       th_ws, beta, logits);

    hipLaunchKernelGGL(col_stats_kernel, dim3((NV + 255) / 256, NL), dim3(256), 0, stream,
                       logits, meanb, rstdb);

    hipLaunchKernelGGL(row_softmax_kernel, dim3(NL * NB), dim3(256), 0, stream,
                       logits, meanb, rstdb, out + OUT_WORD);
}